// BatchedProcrustes_20486994002016
// MI455X (gfx1250) — compile-verified
//
#include <hip/hip_runtime.h>
#include <math.h>

// Batched Procrustes for gfx1250 (MI455X).
// Pipeline: zero ws -> segmented moment reduction -> per-segment 3x3 SVD solve
//           -> WMMA-based affine apply (v_wmma_f32_16x16x4_f32 fast path).

typedef __attribute__((ext_vector_type(2))) float v2f;
typedef __attribute__((ext_vector_type(8))) float v8f;

#define NSEG 4096
#define PTS_PER_THREAD 16

// ---------------------------------------------------------------- zero ws ---
__global__ void zero_acc_kernel(float* __restrict__ acc, int n) {
  int i = blockIdx.x * blockDim.x + threadIdx.x;
  if (i < n) acc[i] = 0.0f;
}

// ------------------------------------------------- segmented moment accum ---
// Per segment b, accumulate 16 floats:
//  [0]=count, [1:4]=sum(src), [4:7]=sum(tgt), [7:16]=sum(src_i * tgt_j)
__global__ __launch_bounds__(256) void accum_kernel(
    const float* __restrict__ src, const float* __restrict__ tgt,
    const int* __restrict__ idx, float* __restrict__ acc, int n) {
  long long tid = (long long)blockIdx.x * blockDim.x + threadIdx.x;
  long long start = tid * PTS_PER_THREAD;
  if (start >= n) return;
  long long end = start + PTS_PER_THREAD;
  if (end > n) end = n;

  float m[16];
#pragma unroll
  for (int k = 0; k < 16; ++k) m[k] = 0.0f;
  int cur = idx[start];

  for (long long p = start; p < end; ++p) {
    int b = idx[p];
    if (b != cur) {  // sorted idx: rare flush on segment boundary
      float* a = acc + (long long)cur * 16;
#pragma unroll
      for (int k = 0; k < 16; ++k) { atomicAdd(a + k, m[k]); m[k] = 0.0f; }
      cur = b;
    }
    float sx = src[p * 3 + 0], sy = src[p * 3 + 1], sz = src[p * 3 + 2];
    float tx = tgt[p * 3 + 0], ty = tgt[p * 3 + 1], tz = tgt[p * 3 + 2];
    m[0] += 1.0f;
    m[1] += sx;  m[2] += sy;  m[3] += sz;
    m[4] += tx;  m[5] += ty;  m[6] += tz;
    m[7]  += sx * tx; m[8]  += sx * ty; m[9]  += sx * tz;
    m[10] += sy * tx; m[11] += sy * ty; m[12] += sy * tz;
    m[13] += sz * tx; m[14] += sz * ty; m[15] += sz * tz;
  }
  float* a = acc + (long long)cur * 16;
#pragma unroll
  for (int k = 0; k < 16; ++k) atomicAdd(a + k, m[k]);
}

// ------------------------------------------------------------ 3x3 helpers ---
__device__ inline float det3(const float M[3][3]) {
  return M[0][0] * (M[1][1] * M[2][2] - M[1][2] * M[2][1])
       - M[0][1] * (M[1][0] * M[2][2] - M[1][2] * M[2][0])
       + M[0][2] * (M[1][0] * M[2][1] - M[1][1] * M[2][0]);
}

__device__ inline void jacobi_rotate(float A[3][3], float V[3][3], int p, int q) {
  float apq = A[p][q];
  if (fabsf(apq) < 1e-30f) return;
  float theta = (A[q][q] - A[p][p]) / (2.0f * apq);
  float t = copysignf(1.0f, theta) / (fabsf(theta) + sqrtf(theta * theta + 1.0f));
  float c = rsqrtf(t * t + 1.0f);
  float s = t * c;
#pragma unroll
  for (int k = 0; k < 3; ++k) {
    float akp = A[k][p], akq = A[k][q];
    A[k][p] = c * akp - s * akq;
    A[k][q] = s * akp + c * akq;
  }
#pragma unroll
  for (int k = 0; k < 3; ++k) {
    float apk = A[p][k], aqk = A[q][k];
    A[p][k] = c * apk - s * aqk;
    A[q][k] = s * apk + c * aqk;
  }
#pragma unroll
  for (int k = 0; k < 3; ++k) {
    float vkp = V[k][p], vkq = V[k][q];
    V[k][p] = c * vkp - s * vkq;
    V[k][q] = s * vkp + c * vkq;
  }
}

// --------------------------------------------------- per-segment SVD solve ---
__global__ __launch_bounds__(256) void solve_kernel(
    const float* __restrict__ acc, float* __restrict__ Rout,
    float* __restrict__ tout, int nseg) {
  int b = blockIdx.x * blockDim.x + threadIdx.x;
  if (b >= nseg) return;
  const float* a = acc + (long long)b * 16;

  float cnt = fmaxf(a[0], 1.0f);
  float inv = 1.0f / cnt;
  float cs[3], ct[3];
#pragma unroll
  for (int i = 0; i < 3; ++i) { cs[i] = a[1 + i] * inv; ct[i] = a[4 + i] * inv; }

  // H = sum(sc tc^T) = M - cnt * cs ct^T
  float H[3][3];
#pragma unroll
  for (int i = 0; i < 3; ++i)
#pragma unroll
    for (int j = 0; j < 3; ++j)
      H[i][j] = a[7 + i * 3 + j] - cnt * cs[i] * ct[j];

  // A = H^T H, symmetric PSD
  float A[3][3];
#pragma unroll
  for (int i = 0; i < 3; ++i)
#pragma unroll
    for (int j = 0; j < 3; ++j)
      A[i][j] = H[0][i] * H[0][j] + H[1][i] * H[1][j] + H[2][i] * H[2][j];

  float V[3][3] = {{1, 0, 0}, {0, 1, 0}, {0, 0, 1}};
  for (int sweep = 0; sweep < 6; ++sweep) {
    jacobi_rotate(A, V, 0, 1);
    jacobi_rotate(A, V, 0, 2);
    jacobi_rotate(A, V, 1, 2);
  }

  // sort eigenpairs descending (so column 2 = smallest singular value)
  float e[3] = {A[0][0], A[1][1], A[2][2]};
  int o[3] = {0, 1, 2};
  if (e[o[0]] < e[o[1]]) { int t = o[0]; o[0] = o[1]; o[1] = t; }
  if (e[o[0]] < e[o[2]]) { int t = o[0]; o[0] = o[2]; o[2] = t; }
  if (e[o[1]] < e[o[2]]) { int t = o[1]; o[1] = o[2]; o[2] = t; }
  float Vs[3][3];
#pragma unroll
  for (int i = 0; i < 3; ++i)
#pragma unroll
    for (int k = 0; k < 3; ++k) Vs[i][k] = V[i][o[k]];

  // U columns: u_k ~ H v_k, orthonormalized; u2 = u0 x u1
  float U[3][3];
  float w[3];
#pragma unroll
  for (int i = 0; i < 3; ++i)
    w[i] = H[i][0] * Vs[0][0] + H[i][1] * Vs[1][0] + H[i][2] * Vs[2][0];
  float n0 = sqrtf(w[0] * w[0] + w[1] * w[1] + w[2] * w[2]);
  if (n0 > 1e-12f) {
#pragma unroll
    for (int i = 0; i < 3; ++i) U[i][0] = w[i] / n0;
  } else {
    U[0][0] = 1.0f; U[1][0] = 0.0f; U[2][0] = 0.0f;
  }
#pragma unroll
  for (int i = 0; i < 3; ++i)
    w[i] = H[i][0] * Vs[0][1] + H[i][1] * Vs[1][1] + H[i][2] * Vs[2][1];
  float dp = U[0][0] * w[0] + U[1][0] * w[1] + U[2][0] * w[2];
#pragma unroll
  for (int i = 0; i < 3; ++i) w[i] -= dp * U[i][0];
  float n1 = sqrtf(w[0] * w[0] + w[1] * w[1] + w[2] * w[2]);
  if (n1 > 1e-12f) {
#pragma unroll
    for (int i = 0; i < 3; ++i) U[i][1] = w[i] / n1;
  } else {
    float ax = fabsf(U[0][0]), ay = fabsf(U[1][0]), az = fabsf(U[2][0]);
    float ev[3] = {0.0f, 0.0f, 0.0f};
    if (ax <= ay && ax <= az) ev[0] = 1.0f;
    else if (ay <= az) ev[1] = 1.0f;
    else ev[2] = 1.0f;
    float d2 = ev[0] * U[0][0] + ev[1] * U[1][0] + ev[2] * U[2][0];
    float q[3];
#pragma unroll
    for (int i = 0; i < 3; ++i) q[i] = ev[i] - d2 * U[i][0];
    float nq = sqrtf(q[0] * q[0] + q[1] * q[1] + q[2] * q[2]);
#pragma unroll
    for (int i = 0; i < 3; ++i) U[i][1] = q[i] / nq;
  }
  U[0][2] = U[1][0] * U[2][1] - U[2][0] * U[1][1];
  U[1][2] = U[2][0] * U[0][1] - U[0][0] * U[2][1];
  U[2][2] = U[0][0] * U[1][1] - U[1][0] * U[0][1];

  // reflection fix: d = sign(det(U) det(V));  R = V diag(1,1,d) U^T
  float d = (det3(U) * det3(Vs)) < 0.0f ? -1.0f : 1.0f;
  float R[3][3];
#pragma unroll
  for (int i = 0; i < 3; ++i)
#pragma unroll
    for (int j = 0; j < 3; ++j)
      R[i][j] = Vs[i][0] * U[j][0] + Vs[i][1] * U[j][1] + d * Vs[i][2] * U[j][2];

  float tv[3];
#pragma unroll
  for (int i = 0; i < 3; ++i)
    tv[i] = ct[i] - (R[i][0] * cs[0] + R[i][1] * cs[1] + R[i][2] * cs[2]);

  float* Rb = Rout + (long long)b * 9;
#pragma unroll
  for (int i = 0; i < 3; ++i)
#pragma unroll
    for (int j = 0; j < 3; ++j) Rb[i * 3 + j] = R[i][j];
  float* tb = tout + (long long)b * 3;
#pragma unroll
  for (int i = 0; i < 3; ++i) tb[i] = tv[i];
}

// ------------------------------------------------------ WMMA affine apply ---
// One wave transforms 16 points: D(16x16, cols 0..2 used) = A(16x4) x B(4x16)
//   A[m][k] = (x,y,z,1) of point m          (16x4 f32 fragment)
//   B[k][n] = R[n][k] (k<3), t[n] (k==3)    (4x16 f32 fragment)
// A-fragment ISA layout (32-bit A 16x4): lanes 0-15 hold K=0,1; lanes 16-31 K=2,3.
// Fast path requires the 16-point chunk to be in one segment (wave-uniform test).
__global__ __launch_bounds__(256) void apply_kernel(
    const float* __restrict__ src, const int* __restrict__ idx,
    const float* __restrict__ Rbuf, const float* __restrict__ tbuf,
    float* __restrict__ out, int n) {
  long long gtid = (long long)blockIdx.x * blockDim.x + threadIdx.x;
  long long wave = gtid >> 5;
  int lane = (int)(gtid & 31);
  long long base = wave * 16;
  if (base >= n) return;  // wave-uniform exit

  long long last = base + 15;
  if (last > (long long)n - 1) last = (long long)n - 1;
  int b0 = idx[base];
  int b1 = idx[last];
  bool fast = (b0 == b1) && (base + 16 <= (long long)n);  // wave-uniform

  if (fast) {
    int b = b0;
    int row = lane & 15;     // A: point row / B: column n
    int khalf = lane >> 4;   // 0 -> K=0,1 ; 1 -> K=2,3

    // A fragment: direct global loads into WMMA lane layout (no shuffles)
    long long p = base + row;
    const float* sp = src + p * 3;
    float a0, a1;
    if (khalf == 0) { a0 = sp[0]; a1 = sp[1]; }
    else            { a0 = sp[2]; a1 = 1.0f; }
    v2f Af = {a0, a1};

    // B fragment: rows of [R^T; t] for this segment (L2/WGP$ resident)
    float bb0 = 0.0f, bb1 = 0.0f;
    if (row < 3) {
      const float* Rr = Rbuf + (long long)b * 9 + row * 3;
      if (khalf == 0) { bb0 = Rr[0]; bb1 = Rr[1]; }
      else            { bb0 = Rr[2]; bb1 = tbuf[(long long)b * 3 + row]; }
    }
    v2f Bf = {bb0, bb1};

    v8f Cf = {};
    // (neg_a, A, neg_b, B, c_mod, C, reuse_a, reuse_b)
    v8f Df = __builtin_amdgcn_wmma_f32_16x16x4_f32(
        false, Af, false, Bf, (short)0, Cf, false, false);

    // D layout: VGPR v -> row m = v (lanes 0-15) / v+8 (lanes 16-31), n = lane%16
    int nn = lane & 15;
    int mofs = (lane >> 4) * 8;
    if (nn < 3) {
#pragma unroll
      for (int v = 0; v < 8; ++v)
        out[(base + mofs + v) * 3 + nn] = Df[v];
    }
  } else {
    // segment boundary inside chunk (or tail): scalar per-point path
    if (lane < 16) {
      long long p = base + lane;
      if (p < n) {
        int b = idx[p];
        const float* Rr = Rbuf + (long long)b * 9;
        const float* tv = tbuf + (long long)b * 3;
        float x = src[p * 3 + 0], y = src[p * 3 + 1], z = src[p * 3 + 2];
#pragma unroll
        for (int i = 0; i < 3; ++i)
          out[p * 3 + i] = Rr[i * 3 + 0] * x + Rr[i * 3 + 1] * y +
                           Rr[i * 3 + 2] * z + tv[i];
      }
    }
  }
}

// ----------------------------------------------------------------- launch ---
extern "C" void kernel_launch(void* const* d_in, const int* in_sizes, int n_in,
                              void* d_out, int out_size, void* d_ws, size_t ws_size,
                              hipStream_t stream) {
  const float* src = (const float*)d_in[0];
  const float* tgt = (const float*)d_in[1];
  const int* idx = (const int*)d_in[2];
  int n = in_sizes[2];          // N_POINTS
  const int nseg = NSEG;        // N_SEGMENTS (fixed by reference)

  float* out = (float*)d_out;                    // aligned [N,3]
  float* Rout = out + (long long)n * 3;          // R [B,3,3]
  float* tout = Rout + (long long)nseg * 9;      // t [B,3]
  float* acc = (float*)d_ws;                     // [B,16] moment accumulators

  // 1) zero accumulators (ws is poisoned by harness; must re-zero every call)
  {
    int total = nseg * 16;
    zero_acc_kernel<<<(total + 255) / 256, 256, 0, stream>>>(acc, total);
  }
  // 2) segmented moment reduction
  {
    long long threads = ((long long)n + PTS_PER_THREAD - 1) / PTS_PER_THREAD;
    int blocks = (int)((threads + 255) / 256);
    accum_kernel<<<blocks, 256, 0, stream>>>(src, tgt, idx, acc, n);
  }
  // 3) per-segment SVD solve -> R, t
  solve_kernel<<<(nseg + 255) / 256, 256, 0, stream>>>(acc, Rout, tout, nseg);
  // 4) apply transform (WMMA fast path, 16 points per wave-chunk)
  {
    long long chunks = ((long long)n + 15) / 16;
    long long threads = chunks * 32;
    int blocks = (int)((threads + 255) / 256);
    apply_kernel<<<blocks, 256, 0, stream>>>(src, idx, Rout, tout, out, n);
  }
}